// TinyFlame_78804059947448
// MI455X (gfx1250) — compile-verified
//
#include <hip/hip_runtime.h>

typedef __attribute__((ext_vector_type(2))) float v2f;
typedef __attribute__((ext_vector_type(8))) float v8f;

#define M_TILE 48   // 16 vertices * 3 components per tile (rows of the GEMM)
#define N_TILE 64   // batch tile
#define K_BLK  40   // K staging block (400 = 10 blocks; posedirs 36 padded to 40)
#define SST    68   // padded LDS stride for the output tile (bank-conflict friendly)

// ---------------------------------------------------------------------------
// Kernel 1: se = concat(shape, expression)  -> (N, 400), coalesced
// ---------------------------------------------------------------------------
__global__ void flame_prep_se(const float* __restrict__ shape,
                              const float* __restrict__ expr,
                              float* __restrict__ se, int N) {
    int idx = blockIdx.x * blockDim.x + threadIdx.x;
    if (idx >= N * 400) return;
    int n = idx / 400, k = idx - n * 400;
    se[idx] = (k < 300) ? shape[n * 300 + k] : expr[n * 100 + (k - 300)];
}

// ---------------------------------------------------------------------------
// Kernel 2: fold J_regressor into the blend bases (done once, O(V) each):
//   Jsd[j,c,s] = sum_v Jr[j,v] * shapedirs[v,c,s]   (15*400 = 6000 outs)
//   Jed[j,c,e] = sum_v Jr[j,v] * eyelids_dirs[v,c,e] (30)
//   Jv [j,c]   = sum_v Jr[j,v] * v_template[v,c]     (15)
// ---------------------------------------------------------------------------
__global__ void flame_jreg(const float* __restrict__ Jr,
                           const float* __restrict__ shapedirs,
                           const float* __restrict__ eyedirs,
                           const float* __restrict__ vtemp,
                           float* __restrict__ Jsd, float* __restrict__ Jed,
                           float* __restrict__ Jv, int V) {
    int t = blockIdx.x * blockDim.x + threadIdx.x;
    if (t < 6000) {
        int jc = t / 400, s = t - jc * 400;
        int j = jc / 3, c = jc - j * 3;
        const float* jr = Jr + (size_t)j * V;
        float acc = 0.f;
        for (int v = 0; v < V; ++v)
            acc += jr[v] * shapedirs[((size_t)(v * 3 + c)) * 400 + s];
        Jsd[t] = acc;
    } else if (t < 6030) {
        int q = t - 6000;
        int jc = q >> 1, e = q & 1;
        int j = jc / 3, c = jc - j * 3;
        const float* jr = Jr + (size_t)j * V;
        float acc = 0.f;
        for (int v = 0; v < V; ++v)
            acc += jr[v] * eyedirs[(size_t)(v * 3 + c) * 2 + e];
        Jed[q] = acc;
    } else if (t < 6045) {
        int q = t - 6030;
        int j = q / 3, c = q - j * 3;
        const float* jr = Jr + (size_t)j * V;
        float acc = 0.f;
        for (int v = 0; v < V; ++v)
            acc += jr[v] * vtemp[v * 3 + c];
        Jv[q] = acc;
    }
}

// ---------------------------------------------------------------------------
// Kernel 3: per-batch joints + kinematic chain + skinning matrices A (N,5,16)
//           and pose features (N,36). One thread per batch element.
// ---------------------------------------------------------------------------
__global__ void flame_chain(const float* __restrict__ se,
                            const float* __restrict__ pose,
                            const float* __restrict__ eyelids,
                            const float* __restrict__ Jsd,
                            const float* __restrict__ Jed,
                            const float* __restrict__ Jv,
                            float* __restrict__ Amat,
                            float* __restrict__ pfeat, int N) {
    int n = blockIdx.x * blockDim.x + threadIdx.x;
    if (n >= N) return;
    const float* se_n = se + (size_t)n * 400;
    const float* pn = pose + (size_t)n * 45;
    float ey0 = eyelids[n * 2], ey1 = eyelids[n * 2 + 1];

    // rest-pose joints (5 joints x 3 comps)
    float jl[15];
    for (int jc = 0; jc < 15; ++jc) {
        float acc = Jv[jc] + Jed[jc * 2] * ey0 + Jed[jc * 2 + 1] * ey1;
        const float* row = Jsd + jc * 400;
        for (int s = 0; s < 400; ++s) acc += row[s] * se_n[s];
        jl[jc] = acc;
    }

    const int par[5] = {-1, 0, 1, 1, 1};
    // compose chain:  T = [R|t]; Tp[j] = Tp[parent] * Tl[j]
    float Rp[5][9], tp[5][3];
    for (int j = 0; j < 5; ++j) {
        float Rl[9], tl[3];
        for (int q = 0; q < 9; ++q) Rl[q] = pn[j * 9 + q];
        int p = par[j];
        for (int c = 0; c < 3; ++c)
            tl[c] = (j == 0) ? jl[c] : (jl[j * 3 + c] - jl[p * 3 + c]);
        if (j == 0) {
            for (int q = 0; q < 9; ++q) Rp[0][q] = Rl[q];
            for (int c = 0; c < 3; ++c) tp[0][c] = tl[c];
        } else {
            for (int r = 0; r < 3; ++r) {
                for (int c = 0; c < 3; ++c) {
                    Rp[j][r * 3 + c] = Rp[p][r * 3 + 0] * Rl[0 * 3 + c] +
                                       Rp[p][r * 3 + 1] * Rl[1 * 3 + c] +
                                       Rp[p][r * 3 + 2] * Rl[2 * 3 + c];
                }
                tp[j][r] = Rp[p][r * 3 + 0] * tl[0] + Rp[p][r * 3 + 1] * tl[1] +
                           Rp[p][r * 3 + 2] * tl[2] + tp[p][r];
            }
        }
    }
    // skinning matrices A = T_posed - pad(T_posed @ [joint;0])
    float* A = Amat + (size_t)n * 80;
    for (int j = 0; j < 5; ++j) {
        for (int i = 0; i < 3; ++i) {
            float corr = Rp[j][i * 3 + 0] * jl[j * 3 + 0] +
                         Rp[j][i * 3 + 1] * jl[j * 3 + 1] +
                         Rp[j][i * 3 + 2] * jl[j * 3 + 2];
            A[j * 16 + i * 4 + 0] = Rp[j][i * 3 + 0];
            A[j * 16 + i * 4 + 1] = Rp[j][i * 3 + 1];
            A[j * 16 + i * 4 + 2] = Rp[j][i * 3 + 2];
            A[j * 16 + i * 4 + 3] = tp[j][i] - corr;
        }
        A[j * 16 + 12] = 0.f; A[j * 16 + 13] = 0.f;
        A[j * 16 + 14] = 0.f; A[j * 16 + 15] = 1.f;
    }
    // pose features: (R[1:] - I) flattened -> 36
    float* pf = pfeat + (size_t)n * 36;
    for (int jj = 0; jj < 4; ++jj)
        for (int r = 0; r < 3; ++r)
            for (int c = 0; c < 3; ++c)
                pf[jj * 9 + r * 3 + c] =
                    pn[(jj + 1) * 9 + r * 3 + c] - ((r == c) ? 1.f : 0.f);
}

// ---------------------------------------------------------------------------
// Kernel 4: fused WMMA GEMM (shapedirs K=400 + posedirs K=36) + template /
// eyelid epilogue + linear blend skinning. 128 threads = 4 wave32 per block.
// Tile: 48 rows (16 vertices x 3 comps) x 64 batch columns.
// ---------------------------------------------------------------------------
__device__ inline void epilogue_store(v8f acc, int fbase, int M0, int MR,
                                      int lane, int n_loc,
                                      const float* __restrict__ vt,
                                      const float* __restrict__ ed,
                                      float ey0, float ey1, float* S) {
    const int half8 = (lane >> 4) * 8;
#pragma unroll
    for (int i = 0; i < 8; ++i) {
        int m_local = fbase + i + half8;     // C/D layout: M = i + 8*(lane>=16)
        int r = M0 + m_local;
        float val = acc[i];
        if (r < MR)
            val += vt[r] + ed[2 * r] * ey0 + ed[2 * r + 1] * ey1;
        S[m_local * SST + n_loc] = val;
    }
}

__global__ __launch_bounds__(128)
void flame_main(const float* __restrict__ se,          // (N,400) ws
                const float* __restrict__ pfeat,       // (N,36)  ws
                const float* __restrict__ Amat,        // (N,80)  ws
                const float* __restrict__ shapedirs,   // (V*3,400)
                const float* __restrict__ posedirs,    // (V*3,36)
                const float* __restrict__ v_template,  // (V*3)
                const float* __restrict__ eyedirs,     // (V*3,2)
                const float* __restrict__ eyelids,     // (N,2)
                const float* __restrict__ weights,     // (V,5)
                const float* __restrict__ translation, // (N,3)
                float* __restrict__ out,               // (N,V,3)
                int V, int N) {
    __shared__ __align__(16) float Ald[M_TILE * K_BLK];   // A tile (rows x K)
    __shared__ __align__(16) float Bld[N_TILE * K_BLK];   // B tile (cols x K)
    __shared__ __align__(16) float S[M_TILE * SST];       // GEMM result tile
    __shared__ __align__(16) float Aw[N_TILE * 80];       // skinning mats
    __shared__ float Wv[16 * 5];                          // vertex weights
    __shared__ float Tr[N_TILE * 3];                      // translations

    const int t = threadIdx.x;
    const int lane = t & 31;
    const int wid = t >> 5;          // 0..3 (wave32)
    const int l15 = lane & 15;
    const int kh = (lane >> 4) * 2;  // A/B frag K offset: lanes 16-31 hold K+2,K+3
    const int wn0 = wid * 16;        // this wave's N sub-tile

    const int MR = V * 3;
    const int M0 = blockIdx.x * M_TILE;
    const int N0 = blockIdx.y * N_TILE;   // N assumed multiple of 64 (N=1024)
    const int V0 = M0 / 3;

    // stage per-tile skinning data (A matrices, weights, translations)
    for (int q = t; q < N_TILE * 80; q += 128) Aw[q] = Amat[(size_t)N0 * 80 + q];
    for (int q = t; q < N_TILE * 3; q += 128) Tr[q] = translation[(size_t)N0 * 3 + q];
    if (t < 80) {
        int v = t / 5;
        Wv[t] = (V0 + v < V) ? weights[(size_t)(V0 + v) * 5 + (t - v * 5)] : 0.f;
    }

    v8f acc0 = {}, acc1 = {}, acc2 = {};

    // 10 blocks of shapedirs (K=400) + 1 zero-padded block of posedirs (K=36)
    for (int blk = 0; blk < 11; ++blk) {
        const bool shp = (blk < 10);
        const int k0 = shp ? blk * K_BLK : 0;
        const int kval = shp ? K_BLK : 36;  // valid K in this block

        __syncthreads();  // previous block's LDS reads done

        // load A tile: rows [M0,M0+48), cols [k0,k0+40)
        for (int q = t; q < M_TILE * K_BLK; q += 128) {
            int m = q / K_BLK, k = q - m * K_BLK;
            int r = M0 + m;
            float val = 0.f;
            if (r < MR && k < kval)
                val = shp ? shapedirs[(size_t)r * 400 + k0 + k]
                          : posedirs[(size_t)r * 36 + k];
            Ald[q] = val;
        }
        // load B tile: cols (batch) [N0,N0+64), rows (K) [k0,k0+40); K-contig in LDS
        for (int q = t; q < N_TILE * K_BLK; q += 128) {
            int nn = q / K_BLK, k = q - nn * K_BLK;
            float val = 0.f;
            if (k < kval)
                val = shp ? se[(size_t)(N0 + nn) * 400 + k0 + k]
                          : pfeat[(size_t)(N0 + nn) * 36 + k];
            Bld[q] = val;
        }
        // hint: prefetch next shapedirs block
        if (blk < 9) {
            int r = M0 + (t % M_TILE);
            if (r < MR) __builtin_prefetch(&shapedirs[(size_t)r * 400 + k0 + K_BLK], 0, 3);
        }
        __syncthreads();

        // 10 WMMA K-steps of 4; 3 M-fragments per wave
#pragma unroll
        for (int ks = 0; ks < K_BLK; ks += 4) {
            v2f b  = *(const v2f*)&Bld[(wn0 + l15) * K_BLK + ks + kh];
            v2f a0 = *(const v2f*)&Ald[(0  + l15) * K_BLK + ks + kh];
            v2f a1 = *(const v2f*)&Ald[(16 + l15) * K_BLK + ks + kh];
            v2f a2 = *(const v2f*)&Ald[(32 + l15) * K_BLK + ks + kh];
            acc0 = __builtin_amdgcn_wmma_f32_16x16x4_f32(false, a0, false, b,
                                                         (short)0, acc0, false, false);
            acc1 = __builtin_amdgcn_wmma_f32_16x16x4_f32(false, a1, false, b,
                                                         (short)0, acc1, false, false);
            acc2 = __builtin_amdgcn_wmma_f32_16x16x4_f32(false, a2, false, b,
                                                         (short)0, acc2, false, false);
        }
    }

    // epilogue: + v_template + eyelid dirs, spill tile to LDS
    const int n_loc = wn0 + l15;
    const int n_g = N0 + n_loc;
    const float ey0 = eyelids[n_g * 2];
    const float ey1 = eyelids[n_g * 2 + 1];
    epilogue_store(acc0, 0,  M0, MR, lane, n_loc, v_template, eyedirs, ey0, ey1, S);
    epilogue_store(acc1, 16, M0, MR, lane, n_loc, v_template, eyedirs, ey0, ey1, S);
    epilogue_store(acc2, 32, M0, MR, lane, n_loc, v_template, eyedirs, ey0, ey1, S);
    __syncthreads();

    // linear blend skinning over the 16x64 (vertex, batch) tile
    const int nv = min(16, V - V0);
    for (int p = t; p < 16 * N_TILE; p += 128) {
        int v = p & 15;
        int nl = p >> 4;
        if (v >= nv) continue;
        const float* An = &Aw[nl * 80];
        float w0 = Wv[v * 5 + 0], w1 = Wv[v * 5 + 1], w2 = Wv[v * 5 + 2];
        float w3 = Wv[v * 5 + 3], w4 = Wv[v * 5 + 4];
        float x = S[(3 * v + 0) * SST + nl];
        float y = S[(3 * v + 1) * SST + nl];
        float z = S[(3 * v + 2) * SST + nl];
        float o[3];
#pragma unroll
        for (int i = 0; i < 3; ++i) {
            float T0 = w0 * An[0 * 16 + i * 4 + 0] + w1 * An[1 * 16 + i * 4 + 0] +
                       w2 * An[2 * 16 + i * 4 + 0] + w3 * An[3 * 16 + i * 4 + 0] +
                       w4 * An[4 * 16 + i * 4 + 0];
            float T1 = w0 * An[0 * 16 + i * 4 + 1] + w1 * An[1 * 16 + i * 4 + 1] +
                       w2 * An[2 * 16 + i * 4 + 1] + w3 * An[3 * 16 + i * 4 + 1] +
                       w4 * An[4 * 16 + i * 4 + 1];
            float T2 = w0 * An[0 * 16 + i * 4 + 2] + w1 * An[1 * 16 + i * 4 + 2] +
                       w2 * An[2 * 16 + i * 4 + 2] + w3 * An[3 * 16 + i * 4 + 2] +
                       w4 * An[4 * 16 + i * 4 + 2];
            float T3 = w0 * An[0 * 16 + i * 4 + 3] + w1 * An[1 * 16 + i * 4 + 3] +
                       w2 * An[2 * 16 + i * 4 + 3] + w3 * An[3 * 16 + i * 4 + 3] +
                       w4 * An[4 * 16 + i * 4 + 3];
            o[i] = T0 * x + T1 * y + T2 * z + T3 + Tr[nl * 3 + i];
        }
        size_t ob = ((size_t)(N0 + nl) * V + (V0 + v)) * 3;
        out[ob + 0] = o[0];
        out[ob + 1] = o[1];
        out[ob + 2] = o[2];
    }
}

// ---------------------------------------------------------------------------
extern "C" void kernel_launch(void* const* d_in, const int* in_sizes, int n_in,
                              void* d_out, int out_size, void* d_ws, size_t ws_size,
                              hipStream_t stream) {
    const float* shape       = (const float*)d_in[0];   // (N,300)
    const float* expression  = (const float*)d_in[1];   // (N,100)
    const float* pose        = (const float*)d_in[2];   // (N,5,3,3)
    const float* translation = (const float*)d_in[3];   // (N,3)
    const float* eyelids     = (const float*)d_in[4];   // (N,2)
    const float* v_template  = (const float*)d_in[5];   // (V,3)
    const float* shapedirs   = (const float*)d_in[6];   // (V,3,400)
    const float* posedirs    = (const float*)d_in[7];   // (V,3,36)
    const float* eyedirs     = (const float*)d_in[8];   // (V,3,2)
    const float* J_regressor = (const float*)d_in[9];   // (5,V)
    const float* weights     = (const float*)d_in[10];  // (V,5)
    float* out = (float*)d_out;

    const int N = in_sizes[0] / 300;
    const int V = in_sizes[5] / 3;

    // workspace layout (floats)
    float* ws   = (float*)d_ws;
    float* Jsd  = ws;                       // 6000
    float* Jed  = ws + 6000;                // 30
    float* Jv   = ws + 6030;                // 15 (+3 pad)
    float* se   = ws + 6048;                // N*400
    float* pf   = se + (size_t)N * 400;     // N*36
    float* Amat = pf + (size_t)N * 36;      // N*80

    // 1) concat shape|expression
    {
        int tot = N * 400;
        flame_prep_se<<<(tot + 255) / 256, 256, 0, stream>>>(shape, expression, se, N);
    }
    // 2) fold J_regressor into blend bases
    flame_jreg<<<(6045 + 255) / 256, 256, 0, stream>>>(J_regressor, shapedirs,
                                                       eyedirs, v_template,
                                                       Jsd, Jed, Jv, V);
    // 3) per-batch kinematics
    flame_chain<<<(N + 127) / 128, 128, 0, stream>>>(se, pose, eyelids, Jsd, Jed,
                                                     Jv, Amat, pf, N);
    // 4) fused WMMA GEMM + skinning
    dim3 grid((V * 3 + M_TILE - 1) / M_TILE, N / N_TILE);
    flame_main<<<grid, 128, 0, stream>>>(se, pf, Amat, shapedirs, posedirs,
                                         v_template, eyedirs, eyelids, weights,
                                         translation, out, V, N);
}